// FMRegNet_52621939310980
// MI455X (gfx1250) — compile-verified
//
#include <hip/hip_runtime.h>

// ---------------------------------------------------------------------------
// FMRegNet forward: P = Phi^T F (fp32 WMMA, LDS-tiled), Gram mats, 2048 solves.
// Target: gfx1250 (CDNA5, wave32, WMMA f32 16x16x4, 320KB LDS/WGP).
// ---------------------------------------------------------------------------

typedef __attribute__((ext_vector_type(2))) float v2f;
typedef __attribute__((ext_vector_type(8))) float v8f;

#define NV     8000
#define FEAT   256
#define NEIG   128
#define NMESH  32
#define NB     16
#define LAM    0.001f

__device__ __forceinline__ v8f wmma_f32(v2f a, v2f b, v8f c) {
    return __builtin_amdgcn_wmma_f32_16x16x4_f32(
        false, a, false, b, (short)0, c, false, false);
}

// ---------------------------------------------------------------------------
// Kernel 1: P[m] (128x256) = Phi[m]^T (128x8000) @ F[m] (8000x256)
//
// Block tile 64(M)x128(N), K-chunk 16, double-buffered LDS panels stored
// K-innermost (transposed) with pad 18 so each WMMA fragment is a single
// bank-conflict-free ds_load_b64.  8 waves as 2x4, each computing 32x32
// (2x2 accumulators) -> 1 LDS load per WMMA.
// ---------------------------------------------------------------------------
__global__ __launch_bounds__(256) void fm_proj_gemm(
    const float* __restrict__ evecs,   // (32*8000, 128)
    const float* __restrict__ feats,   // (32*8000, 256)
    float* __restrict__ P)             // (32, 128, 256)
{
    constexpr int PAD = 18;            // 72B row stride: b64-aligned, no conflicts
    __shared__ float As[2][64][PAD];   // As[buf][m][k]  (Phi^T tile)
    __shared__ float Bs[2][128][PAD];  // Bs[buf][n][k]  (F tile)

    const int tid  = threadIdx.x;
    const int lane = tid & 31;
    const int wave = tid >> 5;
    const int l    = lane & 15;
    const int half = lane >> 4;
    const int mesh = blockIdx.z;
    const int i_base = blockIdx.y * 64;     // M offset in neig
    const int j_base = blockIdx.x * 128;    // N offset in feat
    const int rm = (wave >> 2) * 32;        // wave tile row in block tile
    const int rn = (wave & 3) * 32;         // wave tile col in block tile

    const float* Phi = evecs + (size_t)mesh * NV * NEIG;
    const float* Fm  = feats + (size_t)mesh * NV * FEAT;

    // staging: A panel 16x64 (4 floats/thread), B panel 16x128 (8 floats/thread)
    const int ak = tid >> 4;            // k row 0..15
    const int am = (tid & 15) * 4;      // m col 0..60
    const int bk = tid >> 5;            // k row 0..7 (second half +8)
    const int bn = (tid & 31) * 4;      // n col 0..124

    float4 ra, rb0, rb1;

    auto gload = [&](int kc) {
        ra  = *(const float4*)(Phi + (size_t)(kc + ak)     * NEIG + i_base + am);
        rb0 = *(const float4*)(Fm  + (size_t)(kc + bk)     * FEAT + j_base + bn);
        rb1 = *(const float4*)(Fm  + (size_t)(kc + bk + 8) * FEAT + j_base + bn);
        if (kc + 16 < NV) {  // warm L2 one chunk ahead
            __builtin_prefetch(Phi + (size_t)(kc + 16 + ak) * NEIG + i_base + am, 0, 1);
            __builtin_prefetch(Fm  + (size_t)(kc + 16 + bk) * FEAT + j_base + bn, 0, 1);
        }
    };
    auto sstore = [&](int buf) {
        As[buf][am + 0][ak] = ra.x;  As[buf][am + 1][ak] = ra.y;
        As[buf][am + 2][ak] = ra.z;  As[buf][am + 3][ak] = ra.w;
        Bs[buf][bn + 0][bk] = rb0.x; Bs[buf][bn + 1][bk] = rb0.y;
        Bs[buf][bn + 2][bk] = rb0.z; Bs[buf][bn + 3][bk] = rb0.w;
        Bs[buf][bn + 0][bk + 8] = rb1.x; Bs[buf][bn + 1][bk + 8] = rb1.y;
        Bs[buf][bn + 2][bk + 8] = rb1.z; Bs[buf][bn + 3][bk + 8] = rb1.w;
    };

    v8f acc00 = {}, acc01 = {}, acc10 = {}, acc11 = {};

    gload(0);
    sstore(0);
    __syncthreads();

    constexpr int NCHUNK = NV / 16;   // 500
    for (int c = 0; c < NCHUNK; ++c) {
        const int buf = c & 1;
        if (c + 1 < NCHUNK) gload((c + 1) * 16);

#pragma unroll
        for (int kk = 0; kk < 16; kk += 4) {
            const int k = kk + 2 * half;   // even -> b64 aligned
            const v2f a0 = *(const v2f*)&As[buf][rm + l][k];
            const v2f a1 = *(const v2f*)&As[buf][rm + 16 + l][k];
            const v2f b0 = *(const v2f*)&Bs[buf][rn + l][k];
            const v2f b1 = *(const v2f*)&Bs[buf][rn + 16 + l][k];
            acc00 = wmma_f32(a0, b0, acc00);
            acc01 = wmma_f32(a0, b1, acc01);
            acc10 = wmma_f32(a1, b0, acc10);
            acc11 = wmma_f32(a1, b1, acc11);
        }

        if (c + 1 < NCHUNK) sstore(buf ^ 1);
        __syncthreads();
    }

    // epilogue: C/D layout VGPR j -> row (j + 8*half), col l
    float* Pm = P + (size_t)mesh * NEIG * FEAT;
#pragma unroll
    for (int j = 0; j < 8; ++j) {
        const int r0 = i_base + rm + j + 8 * half;
        const int c0 = j_base + rn + l;
        Pm[(size_t)(r0)      * FEAT + c0]      = acc00[j];
        Pm[(size_t)(r0)      * FEAT + c0 + 16] = acc01[j];
        Pm[(size_t)(r0 + 16) * FEAT + c0]      = acc10[j];
        Pm[(size_t)(r0 + 16) * FEAT + c0 + 16] = acc11[j];
    }
}

// ---------------------------------------------------------------------------
// Kernel 2: Gram matrices.
//   which==0: AAt[b] = A A^T,  A  = P[2b]
//   which==1: BAt[b] = Bm A^T, Bm = P[2b+1]
// C[k,l] = sum_f X[k,f] * Y[l,f]   (M=N=128, K=256) -- tiny, direct global.
// ---------------------------------------------------------------------------
__global__ __launch_bounds__(256) void fm_gram_gemm(
    const float* __restrict__ P,
    float* __restrict__ AAt, float* __restrict__ BAt)
{
    const int lane  = threadIdx.x & 31;
    const int wave  = threadIdx.x >> 5;
    const int l     = lane & 15;
    const int half  = lane >> 4;
    const int b     = blockIdx.z >> 1;
    const int which = blockIdx.z & 1;
    const int k0    = blockIdx.y * 16;
    const int l0    = wave * 16;

    const float* X = P + (size_t)(2 * b + which) * NEIG * FEAT;
    const float* Y = P + (size_t)(2 * b)         * NEIG * FEAT;
    float* out = (which ? BAt : AAt) + (size_t)b * NEIG * NEIG;

    v8f acc = {};
    for (int f0 = 0; f0 < FEAT; f0 += 16) {
#pragma unroll
        for (int ff = 0; ff < 16; ff += 4) {
            const int fb = f0 + ff + 2 * half;
            v2f a, bb;
            a.x  = X[(size_t)(k0 + l) * FEAT + fb];
            a.y  = X[(size_t)(k0 + l) * FEAT + fb + 1];
            bb.x = Y[(size_t)(l0 + l) * FEAT + fb];
            bb.y = Y[(size_t)(l0 + l) * FEAT + fb + 1];
            acc = wmma_f32(a, bb, acc);
        }
    }

#pragma unroll
    for (int j = 0; j < 8; ++j)
        out[(size_t)(k0 + j + 8 * half) * NEIG + l0 + l] = acc[j];
}

// ---------------------------------------------------------------------------
// Kernel 3: one workgroup per (b,i):
//   (AAt[b] + LAM*diag((evx[b,:]-evy[b,i])^2)) x = BAt[b,i,:]
// Augmented 128x129 Gauss-Jordan in LDS (66 KB; CDNA5 allows 320 KB/WG).
// ---------------------------------------------------------------------------
__global__ __launch_bounds__(128) void fm_solve(
    const float* __restrict__ AAt, const float* __restrict__ BAt,
    const float* __restrict__ evals, float* __restrict__ C)
{
    __shared__ float Mm[NEIG * (NEIG + 1)];   // row stride 129, col 128 = rhs
    const int t = threadIdx.x;
    const int i = blockIdx.x;
    const int b = blockIdx.y;

    const float* Ab = AAt + (size_t)b * NEIG * NEIG;
    const float* Bb = BAt + (size_t)b * NEIG * NEIG;

    for (int idx = t; idx < NEIG * NEIG; idx += 128) {
        const int r = idx >> 7, c = idx & 127;
        Mm[r * (NEIG + 1) + c] = Ab[idx];
    }
    __syncthreads();

    {
        const float evx = evals[(size_t)(2 * b)     * NEIG + t];
        const float evy = evals[(size_t)(2 * b + 1) * NEIG + i];
        const float d = evx - evy;
        Mm[t * (NEIG + 1) + t] += LAM * d * d;
        Mm[t * (NEIG + 1) + NEIG] = Bb[(size_t)i * NEIG + t];
    }

    for (int c = 0; c < NEIG; ++c) {
        __syncthreads();
        const float piv = Mm[c * (NEIG + 1) + c];
        if (t != c) {
            const float factor = Mm[t * (NEIG + 1) + c] / piv;
            for (int j = c; j <= NEIG; ++j)
                Mm[t * (NEIG + 1) + j] -= factor * Mm[c * (NEIG + 1) + j];
        }
    }
    __syncthreads();

    const float x = Mm[t * (NEIG + 1) + NEIG] / Mm[t * (NEIG + 1) + t];
    C[((size_t)b * NEIG + i) * NEIG + t] = x;
}

// ---------------------------------------------------------------------------
extern "C" void kernel_launch(void* const* d_in, const int* in_sizes, int n_in,
                              void* d_out, int out_size, void* d_ws, size_t ws_size,
                              hipStream_t stream) {
    const float* features = (const float*)d_in[0];   // (256000, 256)
    const float* evecs    = (const float*)d_in[1];   // (256000, 128)
    const float* evals    = (const float*)d_in[2];   // (4096,)
    (void)in_sizes; (void)n_in; (void)out_size; (void)ws_size;

    float* P   = (float*)d_ws;                         // 4 MB
    float* AAt = P + (size_t)NMESH * NEIG * FEAT;      // 1 MB
    float* BAt = AAt + (size_t)NB * NEIG * NEIG;       // 1 MB
    float* C   = (float*)d_out;                        // (16, 128, 128)

    fm_proj_gemm<<<dim3(2, 2, NMESH), 256, 0, stream>>>(evecs, features, P);
    fm_gram_gemm<<<dim3(1, 8, 2 * NB), 256, 0, stream>>>(P, AAt, BAt);
    fm_solve<<<dim3(NEIG, NB), 128, 0, stream>>>(AAt, BAt, evals, C);
}